// NR_GraphAttention_88862873354909
// MI455X (gfx1250) — compile-verified
//
#include <hip/hip_runtime.h>
#include <hip/hip_bf16.h>

typedef float v2f __attribute__((ext_vector_type(2)));
typedef float v8f __attribute__((ext_vector_type(8)));

// D = A(16x4) x B(4x16) + C(16x16), all fp32, wave32.
// Layout per CDNA5 ISA 7.12.2: A: lane&15 = M, (lane>>4)*2 = K pair; B mirrored with N; C/D row-striped.
static __device__ __forceinline__ v8f wmma4(v2f a, v2f b, v8f c) {
  return __builtin_amdgcn_wmma_f32_16x16x4_f32(false, a, false, b, (short)0, c, false, false);
}

static __device__ __forceinline__ float wave_sum32(float x) {
#pragma unroll
  for (int off = 16; off >= 1; off >>= 1) x += __shfl_xor(x, off, 32);
  return x;
}
static __device__ __forceinline__ float wave_max32(float x) {
#pragma unroll
  for (int off = 16; off >= 1; off >>= 1) x = fmaxf(x, __shfl_xor(x, off, 32));
  return x;
}

#define FEAT 192
#define DIM 64
#define OSTR 196   // padded LDS row stride (192 % 64 == 0 would alias banks)
#define ASTR 68

// --- kernel 1: normalized relation embeddings + per-layer attention logits ---
__global__ __launch_bounds__(256) void prep_rel(const float* __restrict__ rel_emb,
                                                const float* __restrict__ attnk,
                                                float* __restrict__ rn,
                                                float* __restrict__ attdot, int R) {
  int lane = threadIdx.x & 31;
  int r = blockIdx.x * 8 + (threadIdx.x >> 5);
  if (r >= R) return;
  v2f v = *(const v2f*)(rel_emb + (size_t)r * DIM + 2 * lane);
  float s = wave_sum32(v.x * v.x + v.y * v.y);
  float inv = 1.0f / fmaxf(sqrtf(s), 1e-12f);
  v.x *= inv; v.y *= inv;
  *(v2f*)(rn + (size_t)r * DIM + 2 * lane) = v;
#pragma unroll
  for (int l = 0; l < 2; ++l) {
    float d = wave_sum32(v.x * attnk[l * DIM + 2 * lane] + v.y * attnk[l * DIM + 2 * lane + 1]);
    if (lane == 0) attdot[l * R + r] = d;
  }
}

// --- kernel 2: row-l2norm of proxy [64,192] ---
__global__ __launch_bounds__(256) void prep_proxy(const float* __restrict__ proxy,
                                                  float* __restrict__ pn) {
  int lane = threadIdx.x & 31;
  int r = blockIdx.x * 8 + (threadIdx.x >> 5);
  if (r >= 64) return;
  float vals[6], s = 0.f;
#pragma unroll
  for (int t = 0; t < 6; ++t) { vals[t] = proxy[(size_t)r * FEAT + lane + 32 * t]; s += vals[t] * vals[t]; }
  s = wave_sum32(s);
  float inv = 1.0f / fmaxf(sqrtf(s), 1e-12f);
#pragma unroll
  for (int t = 0; t < 6; ++t) pn[(size_t)r * FEAT + lane + 32 * t] = vals[t] * inv;
}

// --- kernel 3: feats0 = tanh(features) into outputs[:, 0:64] ---
__global__ __launch_bounds__(256) void tanh0(const float* __restrict__ feats,
                                             float* __restrict__ outs, int n) {
  int idx = blockIdx.x * blockDim.x + threadIdx.x;
  if (idx >= n * DIM) return;
  int row = idx >> 6, col = idx & 63;
  outs[(size_t)row * FEAT + col] = tanhf(feats[idx]);
}

// --- kernel 4: one GAT layer; one wave per node, DEG<=32 edges ---
__global__ __launch_bounds__(256) void gat_layer(float* __restrict__ outs,
                                                 const float* __restrict__ rn,
                                                 const float* __restrict__ attdot,
                                                 const int* __restrict__ cols,
                                                 const int* __restrict__ relid,
                                                 int layer, int R, int nnodes, int deg) {
  int lane = threadIdx.x & 31;
  int node = blockIdx.x * 8 + (threadIdx.x >> 5);
  if (node >= nnodes) return;
  int ebase = node * deg;
  int src = 0, rl = 0;
  float att = -1e30f;
  if (lane < deg) {
    src = cols[ebase + lane];
    rl  = relid[ebase + lane];
    att = attdot[layer * R + rl];
  }
  float m = wave_max32(att);
  float ex = (lane < deg) ? __expf(att - m) : 0.0f;
  float den = wave_sum32(ex);
  float w = ex / den;

  const int in_off = layer * DIM, out_off = (layer + 1) * DIM;
  float accx = 0.f, accy = 0.f;
  for (int e = 0; e < deg; ++e) {
    int   se = __shfl(src, e, 32);
    int   re = __shfl(rl,  e, 32);
    float we = __shfl(w,   e, 32);
    v2f v = *(const v2f*)(outs + (size_t)se * FEAT + in_off + 2 * lane);
    v2f r = *(const v2f*)(rn + (size_t)re * DIM + 2 * lane);
    float d = wave_sum32(v.x * r.x + v.y * r.y);      // <h, r>
    accx += we * (v.x - 2.f * d * r.x);               // Householder reflection
    accy += we * (v.y - 2.f * d * r.y);
  }
  float* op = outs + (size_t)node * FEAT + out_off + 2 * lane;
  op[0] = tanhf(accx);
  op[1] = tanhf(accy);
}

// --- kernel 5: fused proxy-attention + gate tail; one wave per 16 rows ---
__global__ __launch_bounds__(64) void fuse_tail(const float* __restrict__ outs,
                                                const float* __restrict__ pn,
                                                const float* __restrict__ proxy,
                                                const float* __restrict__ gate_w,
                                                const float* __restrict__ gate_b,
                                                float* __restrict__ out, int ntiles) {
  __shared__ float O_s [2][16 * OSTR];
  __shared__ float PF_s[2][16 * OSTR];
  __shared__ float AT_s[2][16 * ASTR];
  __shared__ float IN_s[2][16];

  int lane = threadIdx.x & 31;
  int wid  = threadIdx.x >> 5;
  int tile = blockIdx.x * 2 + wid;
  if (tile >= ntiles) return;       // wave-uniform exit: EXEC stays all-ones for WMMA
  int row0 = tile * 16;

  float* O  = O_s[wid];
  float* PF = PF_s[wid];
  float* AT = AT_s[wid];
  float* IV = IN_s[wid];

  // stage outputs tile [16,192] (coalesced)
  for (int r = 0; r < 16; ++r)
#pragma unroll
    for (int t = 0; t < 6; ++t)
      O[r * OSTR + lane + 32 * t] = outs[(size_t)(row0 + r) * FEAT + lane + 32 * t];

  // row l2-norms: lanes L / L+16 split each row's 192 cols
  {
    int r = lane & 15, h = lane >> 4;
    float s = 0.f;
    for (int c = h * 96; c < h * 96 + 96; ++c) { float x = O[r * OSTR + c]; s += x * x; }
    s += __shfl_xor(s, 16, 32);
    if (lane < 16) IV[lane] = 1.0f / fmaxf(sqrtf(s), 1e-12f);
  }

  const int fm = lane & 15;          // A row / B col for fragments
  const int kb = (lane >> 4) << 1;   // K pair select
  const int h  = lane >> 4;
  float invm = IV[fm];

  // logits[16,64] = l2norm(o) @ pn^T   (4 N-tiles, K=192)
  v8f L[4];
#pragma unroll
  for (int j = 0; j < 4; ++j) {
    v8f acc;
#pragma unroll
    for (int i = 0; i < 8; ++i) acc[i] = 0.f;
    for (int k0 = 0; k0 < FEAT; k0 += 4) {
      v2f a; a.x = O[fm * OSTR + k0 + kb] * invm; a.y = O[fm * OSTR + k0 + kb + 1] * invm;
      v2f b = *(const v2f*)(pn + (size_t)(j * 16 + fm) * FEAT + k0 + kb);
      acc = wmma4(a, b, acc);
    }
    L[j] = acc;
  }

  // row softmax over 64 proxies, done in D-fragment layout (half-wave reductions)
#pragma unroll
  for (int v = 0; v < 8; ++v) {
    float mx = fmaxf(fmaxf(L[0][v], L[1][v]), fmaxf(L[2][v], L[3][v]));
#pragma unroll
    for (int off = 8; off >= 1; off >>= 1) mx = fmaxf(mx, __shfl_xor(mx, off, 32));
    float e[4], s = 0.f;
#pragma unroll
    for (int j = 0; j < 4; ++j) { e[j] = __expf(L[j][v] - mx); s += e[j]; }
#pragma unroll
    for (int off = 8; off >= 1; off >>= 1) s += __shfl_xor(s, off, 32);
    float invs = 1.0f / s;
    int m = v + 8 * h;
#pragma unroll
    for (int j = 0; j < 4; ++j) AT[m * ASTR + j * 16 + fm] = e[j] * invs;
  }

  // pf = o - att @ proxy   (12 N-tiles, K=64)
  for (int j2 = 0; j2 < 12; ++j2) {
    v8f acc;
#pragma unroll
    for (int i = 0; i < 8; ++i) acc[i] = 0.f;
    int n1 = j2 * 16 + fm;
    for (int k0 = 0; k0 < DIM; k0 += 4) {
      v2f a; a.x = AT[fm * ASTR + k0 + kb]; a.y = AT[fm * ASTR + k0 + kb + 1];
      v2f b; b.x = proxy[(size_t)(k0 + kb) * FEAT + n1];
             b.y = proxy[(size_t)(k0 + kb + 1) * FEAT + n1];
      acc = wmma4(a, b, acc);
    }
#pragma unroll
    for (int v = 0; v < 8; ++v) {
      int m = v + 8 * h;
      PF[m * OSTR + n1] = O[m * OSTR + n1] - acc[v];
    }
  }

  // gate = sigmoid(pf @ gate_w^T + b); out = gate*o + (1-gate)*pf
  for (int j3 = 0; j3 < 12; ++j3) {
    v8f acc;
#pragma unroll
    for (int i = 0; i < 8; ++i) acc[i] = 0.f;
    int n1 = j3 * 16 + fm;
    for (int k0 = 0; k0 < FEAT; k0 += 4) {
      v2f a; a.x = PF[fm * OSTR + k0 + kb]; a.y = PF[fm * OSTR + k0 + kb + 1];
      v2f b = *(const v2f*)(gate_w + (size_t)n1 * FEAT + k0 + kb);
      acc = wmma4(a, b, acc);
    }
    float bias = gate_b[n1];
#pragma unroll
    for (int v = 0; v < 8; ++v) {
      int m = v + 8 * h;
      float g = 1.0f / (1.0f + __expf(-(acc[v] + bias)));
      float o = O[m * OSTR + n1], p = PF[m * OSTR + n1];
      out[(size_t)(row0 + m) * FEAT + n1] = g * o + (1.0f - g) * p;
    }
  }
}

extern "C" void kernel_launch(void* const* d_in, const int* in_sizes, int n_in,
                              void* d_out, int out_size, void* d_ws, size_t ws_size,
                              hipStream_t stream) {
  (void)n_in; (void)out_size; (void)ws_size;
  const float* features = (const float*)d_in[0];
  const float* rel_emb  = (const float*)d_in[1];
  const float* proxy    = (const float*)d_in[2];
  const float* gate_w   = (const float*)d_in[3];
  const float* gate_b   = (const float*)d_in[4];
  const float* attnk    = (const float*)d_in[5];
  const int*   adj      = (const int*)d_in[6];
  const int*   r_index  = (const int*)d_in[7];
  // r_val (d_in[8]) cancels inside l2norm (positive scalar per edge)

  const int E = in_sizes[8];
  const int N = in_sizes[0] / DIM;
  const int R = in_sizes[1] / DIM;
  const int deg = E / N;

  char* ws = (char*)d_ws;
  float* outs   = (float*)ws; ws += (size_t)N * FEAT * sizeof(float);
  float* rn     = (float*)ws; ws += (size_t)R * DIM * sizeof(float);
  float* attdot = (float*)ws; ws += (size_t)2 * R * sizeof(float);
  float* pn     = (float*)ws; ws += (size_t)64 * FEAT * sizeof(float);

  const int* cols  = adj + E;      // adj[1] = neighbor (src) ids
  const int* relid = r_index + E;  // r_index[1] = relation id per edge

  prep_rel  <<<(R + 7) / 8, 256, 0, stream>>>(rel_emb, attnk, rn, attdot, R);
  prep_proxy<<<8, 256, 0, stream>>>(proxy, pn);
  tanh0     <<<(N * DIM + 255) / 256, 256, 0, stream>>>(features, outs, N);
  gat_layer <<<(N + 7) / 8, 256, 0, stream>>>(outs, rn, attdot, cols, relid, 0, R, N, deg);
  gat_layer <<<(N + 7) / 8, 256, 0, stream>>>(outs, rn, attdot, cols, relid, 1, R, N, deg);

  int ntiles = (N + 15) / 16;
  fuse_tail <<<(ntiles + 1) / 2, 64, 0, stream>>>(outs, pn, proxy, gate_w, gate_b,
                                                  (float*)d_out, ntiles);
}